// SimpleBlock_89721866814096
// MI455X (gfx1250) — compile-verified
//
#include <hip/hip_runtime.h>

#define IN_DIM 64
#define OUT_DIM 128
#define K_POINTS 15
#define KDIM (K_POINTS * IN_DIM)   // 960
#define KP_EXTENT 1.2f
#define BN_EPS 1e-5f
#define NEG_SLOPE 0.1f

typedef __attribute__((ext_vector_type(2))) float v2f;
typedef __attribute__((ext_vector_type(8))) float v8f;

// ---------------------------------------------------------------------------
// Kernel 0: zero workspace region (agg + column-sum accumulators)
// ---------------------------------------------------------------------------
__global__ void zero_f(float* __restrict__ p, long n) {
    long i = (long)blockIdx.x * blockDim.x + threadIdx.x;
    long stride = (long)gridDim.x * blockDim.x;
    long n4 = n >> 2;
    for (long j = i; j < n4; j += stride) {
        *(float4*)(p + 4 * j) = make_float4(0.f, 0.f, 0.f, 0.f);
    }
    long tail = n & 3;
    if (i < tail) p[4 * n4 + i] = 0.f;
}

// ---------------------------------------------------------------------------
// Kernel 1: KPConv scatter. One wave (32 lanes) per edge; lane L handles
// feature channels L and L+32 (coalesced 32-dword atomic bursts).
// Kernel-point influence w[e,k] is uniform across the wave -> no divergence
// inside the wave; w<=0 (the common case geometrically) skips all atomics.
// ---------------------------------------------------------------------------
__global__ void kpconv_scatter(const float* __restrict__ feats,
                               const float* __restrict__ pos,
                               const int*   __restrict__ src,
                               const int*   __restrict__ dst,
                               const float* __restrict__ kp,
                               float*       __restrict__ agg,
                               int nEdges) {
    int lane = threadIdx.x & 31;
    int wave = threadIdx.x >> 5;
    int e = blockIdx.x * (blockDim.x >> 5) + wave;
    if (e >= nEdges) return;

    int s = src[e];
    int d = dst[e];
    float dx = pos[3 * s + 0] - pos[3 * d + 0];
    float dy = pos[3 * s + 1] - pos[3 * d + 1];
    float dz = pos[3 * s + 2] - pos[3 * d + 2];

    float f0 = feats[(size_t)s * IN_DIM + lane];
    float f1 = feats[(size_t)s * IN_DIM + lane + 32];

    float* base = agg + (size_t)d * KDIM;

    #pragma unroll
    for (int k = 0; k < K_POINTS; ++k) {
        float ax = dx - kp[3 * k + 0];
        float ay = dy - kp[3 * k + 1];
        float az = dz - kp[3 * k + 2];
        float dist = sqrtf(ax * ax + ay * ay + az * az);
        float w = 1.0f - dist * (1.0f / KP_EXTENT);
        if (w > 0.0f) {   // uniform across the wave
            unsafeAtomicAdd(base + k * IN_DIM + lane,      w * f0);
            unsafeAtomicAdd(base + k * IN_DIM + lane + 32, w * f1);
        }
    }
}

// ---------------------------------------------------------------------------
// Kernel 2: out = agg (M x 960) * W (960 x 128) via V_WMMA_F32_16X16X4_F32.
// Block = 256 threads = 8 waves; blockIdx.x selects a 16-row tile, each wave
// owns one 16-column tile (8 x 16 = 128 columns). 240 chained f32 WMMAs.
// A frag (16x4): lane half=lane>>4 -> float2 at agg[row, k + 2*half].
// B frag (4x16): b.x = W[k+2*half, col], b.y = W[k+2*half+1, col].
// D (16x16): element r of the v8f lives at (row = r + 8*half, col = lane&15).
// Epilogue: uniform branch between full-tile fast path (no exec masking) and
// a guarded tail path (only reachable when nNodes % 16 != 0).
// ---------------------------------------------------------------------------
__global__ void kp_gemm_wmma(const float* __restrict__ agg,
                             const float* __restrict__ W,
                             float*       __restrict__ out,
                             int nNodes) {
    int lane = threadIdx.x & 31;
    int wave = threadIdx.x >> 5;          // 0..7 -> column tile
    int m0   = blockIdx.x * 16;
    int n0   = wave * 16;
    int half = lane >> 4;
    int lm   = lane & 15;

    int arow = m0 + lm;
    if (arow >= nNodes) arow = nNodes - 1;   // clamp keeps EXEC all-ones
    const float* aptr = agg + (size_t)arow * KDIM + 2 * half;
    const float* bptr = W + (size_t)(2 * half) * OUT_DIM + n0 + lm;

    v8f c = {};
    #pragma unroll 8
    for (int k = 0; k < KDIM; k += 4) {
        v2f a = *(const v2f*)(aptr + k);
        v2f b;
        b.x = bptr[(size_t)k * OUT_DIM];
        b.y = bptr[(size_t)k * OUT_DIM + OUT_DIM];
        c = __builtin_amdgcn_wmma_f32_16x16x4_f32(
                /*neg_a=*/false, a, /*neg_b=*/false, b,
                /*c_mod=*/(short)0, c, /*reuse_a=*/false, /*reuse_b=*/false);
    }

    float* dbase = out + (size_t)(m0 + 8 * half) * OUT_DIM + n0 + lm;
    if (m0 + 16 <= nNodes) {
        // full tile: straight-line stores, no per-element predication
        #pragma unroll
        for (int r = 0; r < 8; ++r) {
            dbase[(size_t)r * OUT_DIM] = c[r];
        }
    } else {
        #pragma unroll
        for (int r = 0; r < 8; ++r) {
            int row = m0 + r + 8 * half;
            if (row < nNodes) out[(size_t)row * OUT_DIM + n0 + lm] = c[r];
        }
    }
}

// ---------------------------------------------------------------------------
// Kernel 3: per-channel sum and sum-of-squares for batch-stat BatchNorm.
// Thread -> channel (t & 127); row-lanes grid-stride over rows; two atomics
// per thread at the end (tiny).
// ---------------------------------------------------------------------------
__global__ void col_stats(const float* __restrict__ pre,
                          float*       __restrict__ sums,
                          int nNodes) {
    int c = threadIdx.x & (OUT_DIM - 1);
    int rlane = (int)(((long)blockIdx.x * blockDim.x + threadIdx.x) >> 7);
    int nRowLanes = (int)(((long)gridDim.x * blockDim.x) >> 7);
    float s = 0.f, s2 = 0.f;
    for (int r = rlane; r < nNodes; r += nRowLanes) {
        float v = pre[(size_t)r * OUT_DIM + c];
        s  += v;
        s2 += v * v;
    }
    unsafeAtomicAdd(&sums[c], s);
    unsafeAtomicAdd(&sums[OUT_DIM + c], s2);
}

// ---------------------------------------------------------------------------
// Kernel 4: BatchNorm (population var) + LeakyReLU(0.1).
// ---------------------------------------------------------------------------
__global__ void bn_lrelu(const float* __restrict__ pre,
                         const float* __restrict__ sums,
                         const float* __restrict__ gamma,
                         const float* __restrict__ beta,
                         float*       __restrict__ y,
                         int nNodes) {
    long i = (long)blockIdx.x * blockDim.x + threadIdx.x;
    long total = (long)nNodes * OUT_DIM;
    if (i >= total) return;
    int c = (int)(i & (OUT_DIM - 1));
    float invN = 1.0f / (float)nNodes;
    float mean = sums[c] * invN;
    float var  = sums[OUT_DIM + c] * invN - mean * mean;
    float v = (pre[i] - mean) * rsqrtf(var + BN_EPS) * gamma[c] + beta[c];
    y[i] = (v >= 0.f) ? v : NEG_SLOPE * v;
}

// ---------------------------------------------------------------------------
extern "C" void kernel_launch(void* const* d_in, const int* in_sizes, int n_in,
                              void* d_out, int out_size, void* d_ws, size_t ws_size,
                              hipStream_t stream) {
    const float* feats = (const float*)d_in[0];
    const float* pos   = (const float*)d_in[1];
    const int*   src   = (const int*)  d_in[2];
    const int*   dst   = (const int*)  d_in[3];
    const float* kp    = (const float*)d_in[4];
    const float* W     = (const float*)d_in[5];
    const float* gamma = (const float*)d_in[6];
    const float* beta  = (const float*)d_in[7];

    int nNodes = in_sizes[0] / IN_DIM;
    int nEdges = in_sizes[2];

    // workspace layout: [agg: nNodes*960][sums: 256][pre: nNodes*128]
    float* ws   = (float*)d_ws;
    float* agg  = ws;
    float* sums = ws + (size_t)nNodes * KDIM;
    float* pre  = sums + 2 * OUT_DIM;

    // 0) zero agg + sums
    long zeroN = (long)nNodes * KDIM + 2 * OUT_DIM;
    zero_f<<<2048, 256, 0, stream>>>(ws, zeroN);

    // 1) edge scatter: 8 edges (waves) per 256-thread block
    int edgeBlocks = (nEdges + 7) / 8;
    kpconv_scatter<<<edgeBlocks, 256, 0, stream>>>(feats, pos, src, dst, kp, agg, nEdges);

    // 2) WMMA GEMM: one 16-row tile per block, 8 column tiles per block
    int mTiles = (nNodes + 15) / 16;
    kp_gemm_wmma<<<mTiles, 256, 0, stream>>>(agg, W, pre, nNodes);

    // 3) column stats
    col_stats<<<256, 256, 0, stream>>>(pre, sums, nNodes);

    // 4) BN + LeakyReLU -> d_out
    long total = (long)nNodes * OUT_DIM;
    int finBlocks = (int)((total + 255) / 256);
    bn_lrelu<<<finBlocks, 256, 0, stream>>>(pre, sums, gamma, beta, (float*)d_out, nNodes);
}